// HBVMulTDET_3917010174218
// MI455X (gfx1250) — compile-verified
//
#include <hip/hip_runtime.h>
#include <math.h>

#define NSTEP   365
#define NGRID   2000
#define NMULC   16
#define NPAR    14
#define LENF    15
#define GPB     8            // grids per block
#define TPB     (GPB*NMULC)  // 128 threads = 4 wave32
#define CHUNK   73           // 365 = 5 * 73
#define NCHUNK  5
#define NZ      1e-5f

// ---- gfx1250 async global->LDS path (guarded; falls back to sync copy) ----
#if defined(__AMDGCN__) && __has_builtin(__builtin_amdgcn_global_load_async_to_lds_b128)
#define HAVE_ASYNC 1
#else
#define HAVE_ASYNC 0
#endif

#if HAVE_ASYNC
typedef __attribute__((ext_vector_type(4))) int v4i_t;
typedef __attribute__((address_space(1))) v4i_t* global_v4i_ptr;
typedef __attribute__((address_space(3))) v4i_t* lds_v4i_ptr;
#endif

__device__ __forceinline__ void async_wait_all() {
#if HAVE_ASYNC
#if __has_builtin(__builtin_amdgcn_s_wait_asynccnt)
    __builtin_amdgcn_s_wait_asynccnt(0);
#else
    asm volatile("s_wait_asynccnt 0" ::: "memory");
#endif
#endif
    __syncthreads();
}

// Copy one 73-step chunk of forcings (GPB grids x 3 vars) into LDS.
// Row r (one timestep) = GPB*3*4 = 96 bytes = 6 x b128, all 16B aligned.
__device__ __forceinline__ void stage_chunk(const float* __restrict__ x,
                                            int chunk, int g0, float* dst, int tid) {
    const int NLOADS = CHUNK * 6;
    for (int i = tid; i < NLOADS; i += TPB) {
        int r = i / 6, q = i % 6;
        const float* gsrc = x + (size_t)(chunk * CHUNK + r) * (NGRID * 3) + g0 * 3 + q * 4;
        float* ldst = dst + r * (GPB * 3) + q * 4;
#if HAVE_ASYNC
        __builtin_amdgcn_global_load_async_to_lds_b128(
            (global_v4i_ptr)(gsrc),
            (lds_v4i_ptr)(ldst),
            0, 0);
#else
        float4 v = *(const float4*)gsrc;
        *(float4*)ldst = v;
#endif
    }
}

__global__ __launch_bounds__(TPB)
void hbv_scan_route_kernel(const float* __restrict__ x,        // [365,2000,3]
                           const float* __restrict__ params,   // [365,2000,14,16]
                           const float* __restrict__ convp,    // [2000,2]
                           float* __restrict__ out) {          // [365,2000,1]
    __shared__ __align__(16) float xs[2][CHUNK * GPB * 3];  // double-buffered forcings
    __shared__ float qavg_s[NSTEP * GPB];                   // mean discharge per (t, grid)
    __shared__ float uh_s[GPB * LENF];                      // unit hydrograph per grid

    const int tid = threadIdx.x;
    const int m   = tid & (NMULC - 1);   // component index (16 lanes)
    const int gl  = tid >> 4;            // local grid 0..7
    const int g0  = blockIdx.x * GPB;
    const int g   = g0 + gl;

    // ---- static params: last timestep slice, scaled to physical bounds ----
    const float lo[NPAR] = {1.0f, 50.0f, 0.05f, 0.01f, 0.001f, 0.2f, 0.0f,
                            0.0f, -2.5f, 0.5f, 0.0f, 0.0f, 0.3f, 0.0f};
    const float hi[NPAR] = {6.0f, 1000.0f, 0.9f, 0.5f, 0.2f, 1.0f, 10.0f,
                            100.0f, 2.5f, 10.0f, 0.1f, 0.2f, 5.0f, 1.0f};
    float par[NPAR];
    {
        size_t pbase = (size_t)(NSTEP - 1) * NGRID * NPAR * NMULC
                     + (size_t)g * NPAR * NMULC + m;
#pragma unroll
        for (int i = 0; i < NPAR; ++i)
            par[i] = params[pbase + (size_t)i * NMULC] * (hi[i] - lo[i]) + lo[i];
    }
    const float BETA = par[0],  FC   = par[1], K0    = par[2], K1 = par[3];
    const float K2   = par[4],  LP   = par[5], PERCp = par[6], UZL = par[7];
    const float TT   = par[8],  CFMAX = par[9], CFR  = par[10], CWH = par[11];
    const float BETAET = par[12], C = par[13];
    const float invFC = 1.0f / FC;
    const float invLPFC = 1.0f / (LP * FC);
    const float refc = CFR * CFMAX;

    // ---- state ----
    float SNOWPACK = 0.001f, MELTWATER = 0.001f, SM = 0.001f, SUZ = 0.001f, SLZ = 0.001f;

    // prologue: stage chunk 0
    stage_chunk(x, 0, g0, xs[0], tid);
    async_wait_all();

    for (int c = 0; c < NCHUNK; ++c) {
        if (c + 1 < NCHUNK) stage_chunk(x, c + 1, g0, xs[(c + 1) & 1], tid);
        const float* buf = xs[c & 1];
        const int tbase = c * CHUNK;
        for (int r = 0; r < CHUNK; ++r) {
            const float P   = buf[r * (GPB * 3) + gl * 3 + 0];
            const float T   = buf[r * (GPB * 3) + gl * 3 + 1];
            const float PET = buf[r * (GPB * 3) + gl * 3 + 2];

            const float RAIN = (T >= TT) ? P : 0.0f;
            const float SNOW = (T <  TT) ? P : 0.0f;
            SNOWPACK += SNOW;
            float melt = fmaxf(CFMAX * (T - TT), 0.0f);
            melt = fminf(melt, SNOWPACK);
            MELTWATER += melt;
            SNOWPACK = fmaxf(SNOWPACK - melt, NZ);
            float refr = fmaxf(refc * (TT - T), 0.0f);
            refr = fminf(refr, MELTWATER);
            SNOWPACK += refr;
            MELTWATER = fmaxf(MELTWATER - refr, NZ);
            float tosoil = fmaxf(MELTWATER - CWH * SNOWPACK, 0.0f);
            MELTWATER = fmaxf(MELTWATER - tosoil, NZ);

            const float sw = fminf(fmaxf(__powf(SM * invFC, BETA), 0.0f), 1.0f);
            const float recharge = (RAIN + tosoil) * sw;
            SM = SM + RAIN + tosoil - recharge;
            const float excess = fmaxf(SM - FC, 0.0f);
            SM = fmaxf(SM - excess, NZ);
            const float ef = fminf(fmaxf(__powf(SM * invLPFC, BETAET), 0.0f), 1.0f);
            const float ETact = fminf(SM, PET * ef);
            SM = fmaxf(SM - ETact, NZ);
            const float capil = fminf(SLZ, C * SLZ * (1.0f - fminf(SM * invFC, 1.0f)));
            SM  = fmaxf(SM + capil, NZ);
            SLZ = fmaxf(SLZ - capil, NZ);
            SUZ = SUZ + recharge + excess;
            const float PERC = fminf(SUZ, PERCp);
            SUZ -= PERC;
            const float Q0 = K0 * fmaxf(SUZ - UZL, 0.0f);
            SUZ -= Q0;
            const float Q1 = K1 * SUZ;
            SUZ -= Q1;
            SLZ += PERC;
            const float Q2 = K2 * SLZ;
            SLZ -= Q2;

            // mean over the 16 NMUL components: wave32 butterfly within 16-lane group
            float q = Q0 + Q1 + Q2;
            q += __shfl_xor(q, 1);
            q += __shfl_xor(q, 2);
            q += __shfl_xor(q, 4);
            q += __shfl_xor(q, 8);
            if (m == 0) qavg_s[(tbase + r) * GPB + gl] = q * (1.0f / 16.0f);
        }
        async_wait_all();   // next chunk fully resident in LDS before reuse
    }

    // ---- per-grid gamma unit hydrograph (cold path) ----
    if (tid < GPB) {
        const int gg = g0 + tid;
        const float a  = fmaxf(convp[gg * 2 + 0] * 2.9f, 0.0f) + 0.1f;
        const float th = fmaxf(convp[gg * 2 + 1] * 6.5f, 0.0f) + 0.5f;
        const float denom = expf(lgammaf(a)) * powf(th, a);
        float w[LENF];
        float s = 0.0f;
        for (int k = 0; k < LENF; ++k) {
            const float t = (float)k + 0.5f;
            w[k] = powf(t, a - 1.0f) * expf(-t / th) / denom;
            s += w[k];
        }
        const float invs = 1.0f / s;
        for (int k = 0; k < LENF; ++k) uh_s[tid * LENF + k] = w[k] * invs;
    }
    __syncthreads();

    // ---- causal grouped conv1d routing: out[t,g] = sum_k UH[k,g] * Qavg[t-k,g] ----
    for (int i = tid; i < NSTEP * GPB; i += TPB) {
        const int t = i / GPB, gl2 = i % GPB;   // gl2 fastest -> coalesced stores
        const int kmax = (t < LENF - 1) ? t : (LENF - 1);
        float acc = 0.0f;
        for (int k = 0; k <= kmax; ++k)
            acc += uh_s[gl2 * LENF + k] * qavg_s[(t - k) * GPB + gl2];
        out[(size_t)t * NGRID + g0 + gl2] = acc;
    }
}

extern "C" void kernel_launch(void* const* d_in, const int* in_sizes, int n_in,
                              void* d_out, int out_size, void* d_ws, size_t ws_size,
                              hipStream_t stream) {
    const float* x      = (const float*)d_in[0];  // [365,2000,3]
    // d_in[1] = c_hydro_model: unused by the reference computation
    const float* params = (const float*)d_in[2];  // [365,2000,14,16]
    const float* convp  = (const float*)d_in[3];  // [2000,2]
    float* out = (float*)d_out;                   // [365,2000,1]

    dim3 grid(NGRID / GPB);   // 250 blocks
    dim3 block(TPB);          // 128 threads (4 wave32)
    hbv_scan_route_kernel<<<grid, block, 0, stream>>>(x, params, convp, out);
}